// MatrixFactorization_55619826483857
// MI455X (gfx1250) — compile-verified
//
#include <hip/hip_runtime.h>
#include <hip/hip_bf16.h>

typedef __attribute__((ext_vector_type(2))) float v2f;
typedef __attribute__((ext_vector_type(8))) float v8f;

// -------------------------------------------------------------------------
// Kernel 1: fuse per-feature embedding tables with their W slab via WMMA.
//   P[off_j + v, f] = sum_d emb_j[v,d] * W[4j+d, f]   (+ b[f] folded into j==0)
// One wave (32 threads) per (16-row tile, 16-col half) using
// V_WMMA_F32_16X16X4_F32 : A = 16x4 f32 (emb rows), B = 4x16 f32 (W slab).
// Exact f32 math -> identical results to the reference GEMM path.
// -------------------------------------------------------------------------
__global__ __launch_bounds__(32) void mf_fuse_tables_kernel(
    const float* __restrict__ e0, const float* __restrict__ e1,
    const float* __restrict__ e2, const float* __restrict__ e3,
    const float* __restrict__ e4, const float* __restrict__ e5,
    const float* __restrict__ W,  const float* __restrict__ bias,
    float* __restrict__ P)
{
    const int vocab[6] = {7, 24, 2, 100, 12, 31};
    const int off[6]   = {0, 7, 31, 33, 133, 145};
    // 14 row-tiles covering all features (ceil(vocab/16) each)
    const int tileF[14] = {0, 1,1, 2, 3,3,3,3,3,3,3, 4, 5,5};
    const int tileR[14] = {0, 0,1, 0, 0,1,2,3,4,5,6, 0, 0,1};

    const int task = blockIdx.x;      // 0..27
    const int t    = task >> 1;       // tile id
    const int h    = task & 1;        // column half: cols [16h, 16h+16)
    const int j    = tileF[t];
    const int rt   = tileR[t];
    const int vl   = vocab[j];

    const float* emb = (j == 0) ? e0 : (j == 1) ? e1 : (j == 2) ? e2
                     : (j == 3) ? e3 : (j == 4) ? e4 : e5;

    const int lane = threadIdx.x;     // 0..31, EXEC all ones (no divergence yet)
    const int half = lane >> 4;       // 0/1
    const int lm   = lane & 15;
    const int kb   = half * 2;        // K base held by this lane (K = kb, kb+1)

    // ---- A: 16x4 f32. lane%16 = M, VGPR i holds K = kb + i. Branchless mask.
    const int   v    = rt * 16 + lm;
    const float amsk = (v < vl) ? 1.0f : 0.0f;
    const int   vc   = (v < vl) ? v : 0;
    v2f a;
    a.x = emb[vc * 4 + kb]     * amsk;
    a.y = emb[vc * 4 + kb + 1] * amsk;

    // ---- B: 4x16 f32 slab of W. lane%16 = N (col f), VGPR i holds K = kb + i.
    const int f = h * 16 + lm;
    v2f bm;
    bm.x = W[(4 * j + kb)     * 32 + f];
    bm.y = W[(4 * j + kb + 1) * 32 + f];

    // ---- C: bias folded into the dow feature (added once per output row).
    v8f c;
    const float cb = (j == 0) ? bias[f] : 0.0f;
#pragma unroll
    for (int i = 0; i < 8; ++i) c[i] = cb;

    // D = A x B + C   (v_wmma_f32_16x16x4_f32)
    v8f d = __builtin_amdgcn_wmma_f32_16x16x4_f32(
        /*neg_a=*/false, a, /*neg_b=*/false, bm,
        /*c_mod=*/(short)0, c, /*reuse_a=*/false, /*reuse_b=*/false);

    // ---- Store. D layout: VGPR i, lane -> M = i + 8*(lane/16), N = lane%16.
#pragma unroll
    for (int i = 0; i < 8; ++i) {
        const int m  = i + 8 * half;
        const int vr = rt * 16 + m;
        if (vr < vl) P[(off[j] + vr) * 32 + f] = d[i];
    }
}

// -------------------------------------------------------------------------
// Kernel 2: streaming gather + dot. Memory-bound hot loop.
//   out[r] = (sum_j P_j[idx_j[r]]) . item_table[dest[r]]
// P (22.5 KB) staged into LDS; item rows are contiguous 128B chunks that
// live in the 192MB L2 (table is 12.8 MB), so HBM traffic is ~32 B/row.
// -------------------------------------------------------------------------
__global__ __launch_bounds__(256) void mf_dot_kernel(
    const int* __restrict__ dow,  const int* __restrict__ tim,
    const int* __restrict__ sex,  const int* __restrict__ age,
    const int* __restrict__ mon,  const int* __restrict__ day,
    const int* __restrict__ dest, const float* __restrict__ item,
    const float* __restrict__ P,  float* __restrict__ out, int n)
{
    __shared__ float sP[176 * 32];                  // fused tables, 22528 B
    {
        const float4* __restrict__ Pg = (const float4*)P;
        float4* sPv = (float4*)sP;
        for (int i = threadIdx.x; i < 176 * 8; i += blockDim.x) sPv[i] = Pg[i];
    }
    __syncthreads();

    const int r = blockIdx.x * blockDim.x + threadIdx.x;
    if (r >= n) return;

    const float4* p0 = (const float4*)(sP + (0   + dow[r]) * 32);
    const float4* p1 = (const float4*)(sP + (7   + tim[r]) * 32);
    const float4* p2 = (const float4*)(sP + (31  + sex[r]) * 32);
    const float4* p3 = (const float4*)(sP + (33  + age[r]) * 32);
    const float4* p4 = (const float4*)(sP + (133 + mon[r]) * 32);
    const float4* p5 = (const float4*)(sP + (145 + day[r]) * 32);
    const float4* it = (const float4*)(item + (size_t)dest[r] * 32);

    float acc = 0.0f;
#pragma unroll
    for (int c = 0; c < 8; ++c) {
        const float4 A = p0[c], B = p1[c], C = p2[c],
                     D = p3[c], E = p4[c], F = p5[c];
        const float4 iv = it[c];
        const float sx = A.x + B.x + C.x + D.x + E.x + F.x;
        const float sy = A.y + B.y + C.y + D.y + E.y + F.y;
        const float sz = A.z + B.z + C.z + D.z + E.z + F.z;
        const float sw = A.w + B.w + C.w + D.w + E.w + F.w;
        acc = fmaf(sx, iv.x, acc);
        acc = fmaf(sy, iv.y, acc);
        acc = fmaf(sz, iv.z, acc);
        acc = fmaf(sw, iv.w, acc);
    }
    out[r] = acc;
}

// -------------------------------------------------------------------------
extern "C" void kernel_launch(void* const* d_in, const int* in_sizes, int n_in,
                              void* d_out, int out_size, void* d_ws, size_t ws_size,
                              hipStream_t stream)
{
    const int*   dow  = (const int*)  d_in[0];
    const int*   tim  = (const int*)  d_in[1];
    const int*   sex  = (const int*)  d_in[2];
    const int*   age  = (const int*)  d_in[3];
    const int*   mon  = (const int*)  d_in[4];
    const int*   day  = (const int*)  d_in[5];
    const int*   dest = (const int*)  d_in[6];
    const float* e0   = (const float*)d_in[7];   // emb_dow   [7,4]
    const float* e1   = (const float*)d_in[8];   // emb_time  [24,4]
    const float* e2   = (const float*)d_in[9];   // emb_sex   [2,4]
    const float* e3   = (const float*)d_in[10];  // emb_age   [100,4]
    const float* e4   = (const float*)d_in[11];  // emb_month [12,4]
    const float* e5   = (const float*)d_in[12];  // emb_day   [31,4]
    const float* W    = (const float*)d_in[13];  // [24,32]
    const float* bias = (const float*)d_in[14];  // [32]
    const float* item = (const float*)d_in[15];  // [100000,32]
    float*       outp = (float*)d_out;

    const int N = in_sizes[0];
    float* P = (float*)d_ws;                     // 176 x 32 fused table (22528 B)

    // 28 single-wave blocks: 14 row tiles x 2 column halves, one WMMA each.
    mf_fuse_tables_kernel<<<28, 32, 0, stream>>>(e0, e1, e2, e3, e4, e5, W, bias, P);

    const int threads = 256;
    const int blocks  = (N + threads - 1) / threads;
    mf_dot_kernel<<<blocks, threads, 0, stream>>>(dow, tim, sex, age, mon, day,
                                                  dest, item, P, outp, N);
}